// OGNN_16604343566802
// MI455X (gfx1250) — compile-verified
//
#include <hip/hip_runtime.h>
#include <hip/hip_bf16.h>
#include <float.h>
#include <limits.h>

typedef float v2f __attribute__((ext_vector_type(2)));
typedef float v8f __attribute__((ext_vector_type(8)));

#define HID 128
#define BN_EPS 1e-5f
#define LDS_PITCH 132   // floats: 528B rows -> 16B-aligned float4 fill, 4-bank read stride

// ---------------- utility kernels ----------------

__global__ void k_zero_f32(float* __restrict__ p, int n) {
    int i = blockIdx.x * blockDim.x + threadIdx.x;
    int stride = gridDim.x * blockDim.x;
    for (; i < n; i += stride) p[i] = 0.0f;
}

__global__ void k_set_int(int* p, int v) { p[0] = v; }

__global__ void k_copy_f4(float4* __restrict__ dst, const float4* __restrict__ src, int n4) {
    int i = blockIdx.x * blockDim.x + threadIdx.x;
    int stride = gridDim.x * blockDim.x;
    for (; i < n4; i += stride) dst[i] = src[i];
}

// ---------------- graph preprocessing ----------------

__global__ void k_rowmin(const int* __restrict__ row, int E, int* __restrict__ rmin) {
    __shared__ int s;
    if (threadIdx.x == 0) s = INT_MAX;
    __syncthreads();
    int v = INT_MAX;
    int i = blockIdx.x * blockDim.x + threadIdx.x;
    int stride = gridDim.x * blockDim.x;
    for (; i < E; i += stride) v = min(v, row[i]);
    atomicMin(&s, v);
    __syncthreads();
    if (threadIdx.x == 0) atomicMin(rmin, s);
}

__global__ void k_deg(const int* __restrict__ col, int E, float* __restrict__ deg) {
    int i = blockIdx.x * blockDim.x + threadIdx.x;
    int stride = gridDim.x * blockDim.x;
    for (; i < E; i += stride) atomicAdd(&deg[col[i]], 1.0f);
}

__global__ void k_dis(const float* __restrict__ deg, float* __restrict__ dis, int n) {
    int i = blockIdx.x * blockDim.x + threadIdx.x;
    if (i < n) {
        float d = deg[i];
        dis[i] = (d > 0.0f) ? rsqrtf(d) : 0.0f;
    }
}

// ---------------- WMMA fp32 GEMM with LDS-staged B panel ----------------
// out[M,OC] = A[M,K] @ W[OC,K]^T + bias, K == 128.
// Block = 256 threads = 8 waves: 8 M-tiles (16 rows each) x one 64-col N-quad.
// B panel (64 x 128 f32) cooperatively staged in LDS (cols >= OC zero-filled),
// then read as ds_load_b64 fragments: row pitch 132 floats => lane bank delta 4,
// kb-half offset 2 banks => conflict-free. A fragment stays in global with a
// one-step register prefetch. EXEC is all-ones for every WMMA.

__device__ __forceinline__ v8f wmma_f32(v2f a, v2f b, v8f c) {
    return __builtin_amdgcn_wmma_f32_16x16x4_f32(false, a, false, b, (short)0, c,
                                                 false, false);
}

__device__ __forceinline__ void gemm_fill_b(float* bs, const float* __restrict__ W,
                                            int n0, int K, int OC) {
    for (int cidx = threadIdx.x; cidx < 64 * 32; cidx += blockDim.x) {
        int rowl = cidx >> 5;            // local col 0..63
        int koff = (cidx & 31) << 2;     // 0..124
        int gcol = n0 + rowl;
        float4 v4 = make_float4(0.f, 0.f, 0.f, 0.f);
        if (gcol < OC) v4 = *(const float4*)(W + (size_t)gcol * K + koff);
        *(float4*)(&bs[rowl * LDS_PITCH + koff]) = v4;
    }
}

__global__ __launch_bounds__(256)
void k_gemm_wmma(const float* __restrict__ A, const float* __restrict__ W,
                 const float* __restrict__ bias, float* __restrict__ out,
                 int nrows, int K, int OC) {
    __shared__ float bs[64 * LDS_PITCH];
    int nquads = (OC + 63) >> 6;
    int bq = blockIdx.x % nquads;
    int bm = blockIdx.x / nquads;
    int n0 = bq << 6;

    gemm_fill_b(bs, W, n0, K, OC);
    __syncthreads();

    int wave = threadIdx.x >> 5;
    int lane = threadIdx.x & 31;
    int m0 = (bm * 8 + wave) << 4;
    if (m0 >= nrows) return;

    int l15 = lane & 15;
    int kb  = (lane >> 4) << 1;          // 0 or 2
    const float* ap  = A + (size_t)(m0 + l15) * K + kb;
    const float* bs0 = &bs[(l15 +  0) * LDS_PITCH + kb];
    const float* bs1 = &bs[(l15 + 16) * LDS_PITCH + kb];
    const float* bs2 = &bs[(l15 + 32) * LDS_PITCH + kb];
    const float* bs3 = &bs[(l15 + 48) * LDS_PITCH + kb];

    v8f c0 = {}, c1 = {}, c2 = {}, c3 = {};
    v2f a = *(const v2f*)(ap);           // prologue A fragment
#pragma unroll 4
    for (int k = 4; k < K; k += 4) {
        v2f an = *(const v2f*)(ap + k);  // prefetch next A fragment (global)
        v2f b0 = *(const v2f*)(bs0 + k - 4);
        v2f b1 = *(const v2f*)(bs1 + k - 4);
        v2f b2 = *(const v2f*)(bs2 + k - 4);
        v2f b3 = *(const v2f*)(bs3 + k - 4);
        c0 = wmma_f32(a, b0, c0);
        c1 = wmma_f32(a, b1, c1);
        c2 = wmma_f32(a, b2, c2);
        c3 = wmma_f32(a, b3, c3);
        a = an;
    }
    {
        int k = K - 4;
        v2f b0 = *(const v2f*)(bs0 + k);
        v2f b1 = *(const v2f*)(bs1 + k);
        v2f b2 = *(const v2f*)(bs2 + k);
        v2f b3 = *(const v2f*)(bs3 + k);
        c0 = wmma_f32(a, b0, c0);
        c1 = wmma_f32(a, b1, c1);
        c2 = wmma_f32(a, b2, c2);
        c3 = wmma_f32(a, b3, c3);
    }

    int rbase = m0 + ((lane >> 4) << 3);
    int col0 = n0 + l15;
#define STORE_TILE(cc, colj)                                               \
    if ((colj) < OC) {                                                     \
        float bv = bias[colj];                                             \
        _Pragma("unroll")                                                  \
        for (int v = 0; v < 8; ++v)                                        \
            out[(size_t)(rbase + v) * OC + (colj)] = (cc)[v] + bv;         \
    }
    STORE_TILE(c0, col0) STORE_TILE(c1, col0 + 16)
    STORE_TILE(c2, col0 + 32) STORE_TILE(c3, col0 + 48)
#undef STORE_TILE
}

// ---------------- fused head GEMM: out = relu(p0*xX + p1*hX + p2*hA) @ Wp^T + bp ----
// Single N-quad (OC <= 64) => each h element consumed exactly once, so the
// softmax-combine + relu is fused into the A-fragment path (saves a full
// 25.6MB write + read at 23.3 TB/s). B = Wp staged in LDS as above.

__device__ __forceinline__ v2f combine_relu(v2f vx, v2f vh, v2f va,
                                            float e0, float e1, float e2) {
    v2f a;
    a.x = fmaxf(e0 * vx.x + e1 * vh.x + e2 * va.x, 0.0f);
    a.y = fmaxf(e0 * vx.y + e1 * vh.y + e2 * va.y, 0.0f);
    return a;
}

__global__ __launch_bounds__(256)
void k_gemm_head(const float* __restrict__ xX, const float* __restrict__ hX,
                 const float* __restrict__ hA, const float* __restrict__ policy,
                 const float* __restrict__ W, const float* __restrict__ bias,
                 float* __restrict__ out, int nrows, int K, int OC) {
    __shared__ float bs[64 * LDS_PITCH];
    gemm_fill_b(bs, W, 0, K, OC);
    __syncthreads();

    int wave = threadIdx.x >> 5;
    int lane = threadIdx.x & 31;
    int m0 = (blockIdx.x * 8 + wave) << 4;
    if (m0 >= nrows) return;

    float p0 = policy[0], p1 = policy[1], p2 = policy[2];
    float m = fmaxf(p0, fmaxf(p1, p2));
    float e0 = __expf(p0 - m), e1 = __expf(p1 - m), e2 = __expf(p2 - m);
    float inv = 1.0f / (e0 + e1 + e2);
    e0 *= inv; e1 *= inv; e2 *= inv;

    int l15 = lane & 15;
    int kb  = (lane >> 4) << 1;
    size_t abase = (size_t)(m0 + l15) * K + kb;
    const float* ax = xX + abase;
    const float* ah = hX + abase;
    const float* aa = hA + abase;
    const float* bs0 = &bs[(l15 +  0) * LDS_PITCH + kb];
    const float* bs1 = &bs[(l15 + 16) * LDS_PITCH + kb];
    const float* bs2 = &bs[(l15 + 32) * LDS_PITCH + kb];
    const float* bs3 = &bs[(l15 + 48) * LDS_PITCH + kb];

    v8f c0 = {}, c1 = {}, c2 = {}, c3 = {};
    v2f a = combine_relu(*(const v2f*)(ax), *(const v2f*)(ah), *(const v2f*)(aa),
                         e0, e1, e2);
#pragma unroll 4
    for (int k = 4; k < K; k += 4) {
        v2f vx = *(const v2f*)(ax + k);  // prefetch next combined A fragment
        v2f vh = *(const v2f*)(ah + k);
        v2f va = *(const v2f*)(aa + k);
        v2f b0 = *(const v2f*)(bs0 + k - 4);
        v2f b1 = *(const v2f*)(bs1 + k - 4);
        v2f b2 = *(const v2f*)(bs2 + k - 4);
        v2f b3 = *(const v2f*)(bs3 + k - 4);
        c0 = wmma_f32(a, b0, c0);
        c1 = wmma_f32(a, b1, c1);
        c2 = wmma_f32(a, b2, c2);
        c3 = wmma_f32(a, b3, c3);
        a = combine_relu(vx, vh, va, e0, e1, e2);
    }
    {
        int k = K - 4;
        v2f b0 = *(const v2f*)(bs0 + k);
        v2f b1 = *(const v2f*)(bs1 + k);
        v2f b2 = *(const v2f*)(bs2 + k);
        v2f b3 = *(const v2f*)(bs3 + k);
        c0 = wmma_f32(a, b0, c0);
        c1 = wmma_f32(a, b1, c1);
        c2 = wmma_f32(a, b2, c2);
        c3 = wmma_f32(a, b3, c3);
    }

    int rbase = m0 + ((lane >> 4) << 3);
    int col0 = l15;
#define STORE_TILE(cc, colj)                                               \
    if ((colj) < OC) {                                                     \
        float bv = bias[colj];                                             \
        _Pragma("unroll")                                                  \
        for (int v = 0; v < 8; ++v)                                        \
            out[(size_t)(rbase + v) * OC + (colj)] = (cc)[v] + bv;         \
    }
    STORE_TILE(c0, col0) STORE_TILE(c1, col0 + 16)
    STORE_TILE(c2, col0 + 32) STORE_TILE(c3, col0 + 48)
#undef STORE_TILE
}

// ---------------- SpMM scatter (one wave per edge, 4 channels per lane) ----------------
// Edge indices are wave-uniform: force into SGPRs so row/col/dis reads become s_loads.

template <bool NORM>
__global__ void k_spmm(const int* __restrict__ row, const int* __restrict__ col,
                       const int* __restrict__ rmin, const float* __restrict__ dis,
                       const float* __restrict__ hin, float* __restrict__ hout,
                       int E) {
    int e = (blockIdx.x * blockDim.x + threadIdx.x) >> 5;
    if (e >= E) return;
    e = __builtin_amdgcn_readfirstlane(e);
    int lane = threadIdx.x & 31;
    int r = row[e] - rmin[0];
    int c = col[e];
    float w = 1.0f;
    if (NORM) w = dis[r] * dis[c];
    const float4 v = *(const float4*)(hin + (size_t)c * HID + lane * 4);
    float* dst = hout + (size_t)r * HID + lane * 4;
    atomicAdd(dst + 0, w * v.x);
    atomicAdd(dst + 1, w * v.y);
    atomicAdd(dst + 2, w * v.z);
    atomicAdd(dst + 3, w * v.w);
}

// ---------------- BatchNorm statistics (column sums over N rows) ----------------
__global__ void k_stats(const float* __restrict__ t, float* __restrict__ sums,
                        float* __restrict__ sumsq, int nrows) {
    int ch = threadIdx.x;   // blockDim.x == 128
    float s = 0.0f, sq = 0.0f;
    for (int i = blockIdx.x; i < nrows; i += gridDim.x) {
        float v = t[(size_t)i * HID + ch];
        s += v; sq += v * v;
    }
    atomicAdd(&sums[ch], s);
    atomicAdd(&sumsq[ch], sq);
}

// normalize in place + accumulate into hA
__global__ void k_bn(float* __restrict__ t, float* __restrict__ hA,
                     const float* __restrict__ sums, const float* __restrict__ sumsq,
                     const float* __restrict__ gamma, const float* __restrict__ beta,
                     int total, float invN) {
    int i = blockIdx.x * blockDim.x + threadIdx.x;
    int stride = gridDim.x * blockDim.x;
    for (; i < total; i += stride) {
        int ch = i & (HID - 1);
        float mu  = sums[ch] * invN;
        float var = fmaxf(sumsq[ch] * invN - mu * mu, 0.0f);
        float w = (t[i] - mu) * rsqrtf(var + BN_EPS) * gamma[ch] + beta[ch];
        t[i] = w;
        hA[i] += w;
    }
}

// ---------------- host orchestration ----------------

extern "C" void kernel_launch(void* const* d_in, const int* in_sizes, int n_in,
                              void* d_out, int out_size, void* d_ws, size_t ws_size,
                              hipStream_t stream) {
    const float* x      = (const float*)d_in[0];
    const int*   ei     = (const int*)  d_in[1];
    const float* Wx     = (const float*)d_in[2];
    const float* bx     = (const float*)d_in[3];
    const float* wA     = (const float*)d_in[4];
    const float* gammas = (const float*)d_in[5];
    const float* betas  = (const float*)d_in[6];
    const float* policy = (const float*)d_in[7];
    const float* Wp     = (const float*)d_in[8];
    const float* bpb    = (const float*)d_in[9];

    const int N  = in_sizes[0] / HID;
    const int E  = in_sizes[1] / 2;
    const int OC = in_sizes[9];           // 47
    const int* row = ei;
    const int* col = ei + E;

    const size_t NB = (size_t)N * HID;    // elements per feature matrix
    float* B0 = (float*)d_ws;             // xX
    float* B1 = B0 + NB;
    float* B2 = B1 + NB;
    float* B3 = B2 + NB;                  // hA
    float* B4 = B3 + NB;                  // scratch (hA-loop ping buffer)
    float* sums  = B4 + NB;               // [128]
    float* sumsq = sums + HID;            // [128]
    float* deg   = sumsq + HID;           // [N]
    float* dis   = deg + N;               // [N]
    int*   rmin  = (int*)(dis + N);

    const int total = (int)NB;
    const int TB = 256;
    const int gElem  = (total + TB - 1) / TB;
    const int gEdge  = (E + TB - 1) / TB;
    const int gEdgeW = (E * 32 + TB - 1) / TB;     // one wave per edge

    // init: deg=0, hA=0, rmin=INT_MAX
    k_zero_f32<<<(N + TB - 1) / TB, TB, 0, stream>>>(deg, N);
    k_zero_f32<<<gElem, TB, 0, stream>>>(B3, total);
    k_set_int<<<1, 1, 0, stream>>>(rmin, INT_MAX);

    k_rowmin<<<256, TB, 0, stream>>>(row, E, rmin);
    k_deg<<<gEdge, TB, 0, stream>>>(col, E, deg);
    k_dis<<<(N + TB - 1) / TB, TB, 0, stream>>>(deg, dis, N);

    const int mtiles  = (N + 15) / 16;
    const int mblocks = (mtiles + 7) / 8;          // 8 M-tiles per block

    // GEMM1: xX = x @ Wx^T + bx   -> B0
    k_gemm_wmma<<<mblocks * ((HID + 63) / 64), TB, 0, stream>>>(x, Wx, bx, B0,
                                                                N, HID, HID);

    // hX loop: 4x (dst = xX; dst[r] += norm * src[c])   B1/B2 ping-pong -> final in B2
    {
        const float* src = B0;
        for (int it = 0; it < 4; ++it) {
            float* dst = (it & 1) ? B2 : B1;
            k_copy_f4<<<(total / 4 + TB - 1) / TB, TB, 0, stream>>>(
                (float4*)dst, (const float4*)B0, total / 4);
            k_spmm<true><<<gEdgeW, TB, 0, stream>>>(row, col, rmin, dis, src, dst, E);
            src = dst;
        }
    }

    // hA loop: 3x (t = A_ori @ src; BN(t); hA += t)    B1/B4 ping-pong
    {
        const float* src = wA;
        for (int i = 0; i < 3; ++i) {
            float* t = (i & 1) ? B4 : B1;
            k_zero_f32<<<gElem, TB, 0, stream>>>(t, total);
            k_spmm<false><<<gEdgeW, TB, 0, stream>>>(row, col, rmin, dis, src, t, E);
            k_zero_f32<<<1, 256, 0, stream>>>(sums, 2 * HID);
            k_stats<<<512, HID, 0, stream>>>(t, sums, sumsq, N);
            k_bn<<<gElem, TB, 0, stream>>>(t, B3, sums, sumsq,
                                           gammas + i * HID, betas + i * HID,
                                           total, 1.0f / (float)N);
            src = t;
        }
    }

    // head: out = relu(softmax(policy).[xX,hX,hA]) @ Wp^T + bp  (fused combine)
    k_gemm_head<<<mblocks, TB, 0, stream>>>(B0, B2, B3, policy, Wp, bpb,
                                            (float*)d_out, N, HID, OC);
}